// GCN_BCSR_29918742184457
// MI455X (gfx1250) — compile-verified
//
#include <hip/hip_runtime.h>

typedef __attribute__((ext_vector_type(2))) float v2f;
typedef __attribute__((ext_vector_type(8))) float v8f;

#define LN_EPS 1e-5f
#define GCN_N   131072
#define GCN_T   16
#define GCN_BR  (GCN_N / GCN_T)   // 8192 block rows == M tiles
#define GCN_K   8

static __device__ __forceinline__ v8f wmma_f32(v2f a, v2f b, v8f c) {
  // D = A(16x4) * B(4x16) + C(16x16), all fp32, wave32
  return __builtin_amdgcn_wmma_f32_16x16x4_f32(false, a, false, b, (short)0, c, false, false);
}

// ---------------------------------------------------------------------------
// Interleave a [K x Nc] row-major matrix into pair-of-rows layout:
//   Wp[((k>>1)*Nc + n)*2 + (k&1)] = W[k*Nc + n]
// so a WMMA B operand (rows k, k+1 at column n, k even) is one b64 load.
// ---------------------------------------------------------------------------
__global__ void interleave_rows(const float* __restrict__ W, float* __restrict__ Wp,
                                int K, int Nc) {
  int idx = blockIdx.x * blockDim.x + threadIdx.x;
  if (idx >= K * Nc) return;
  int k = idx / Nc, n = idx - k * Nc;
  Wp[(((size_t)(k >> 1) * Nc + n) << 1) + (k & 1)] = W[idx];
}

// ---------------------------------------------------------------------------
// Dense GEMM: Zint = X[M x FIN] @ Wp[FIN x FOUT]  (Wp, Zint pair-interleaved)
// One wave computes a full 16 x FOUT strip (FOUT/16 accumulator tiles).
// ---------------------------------------------------------------------------
template <int FIN, int FOUT>
__launch_bounds__(128) __global__
void gemm_kernel(const float* __restrict__ X, const float* __restrict__ Wp,
                 float* __restrict__ Zint) {
  constexpr int NT = FOUT / 16;
  const int lane = threadIdx.x & 31;
  const int wave = threadIdx.x >> 5;
  const int tile = blockIdx.x * (blockDim.x >> 5) + wave; // 16-row M tile
  const int n  = lane & 15;   // A row index / B,C col index
  const int kk = lane >> 4;   // lane half

  v8f acc[NT];
#pragma unroll
  for (int t = 0; t < NT; ++t)
#pragma unroll
    for (int i = 0; i < 8; ++i) acc[t][i] = 0.0f;

  const float* xrow = X + (size_t)(tile * 16 + n) * FIN;
#pragma unroll 4
  for (int kb = 0; kb < FIN; kb += 4) {
    // A operand: lane holds A[m][kb+2*kk .. +1] (float2, 8B aligned)
    v2f a = *(const v2f*)(xrow + kb + 2 * kk);
    // B operand rows (kb+2*kk, kb+2*kk+1): one b64 from interleaved Wp
    const float* w0 = Wp + ((size_t)((kb + 2 * kk) >> 1) * FOUT) * 2;
#pragma unroll
    for (int t = 0; t < NT; ++t) {
      v2f b = *(const v2f*)(w0 + (t * 16 + n) * 2);
      acc[t] = wmma_f32(a, b, acc[t]);
    }
  }

  // C/D layout: VGPR i -> row (tile*16 + 8*kk + i), col (t*16 + n).
  // Adjacent i form an even/odd row pair -> b64 store into interleaved Zint.
  float* zbase = Zint + (size_t)((tile * 16 + 8 * kk) >> 1) * FOUT * 2;
#pragma unroll
  for (int t = 0; t < NT; ++t)
#pragma unroll
    for (int i2 = 0; i2 < 4; ++i2) {
      v2f p;
      p.x = acc[t][2 * i2];
      p.y = acc[t][2 * i2 + 1];
      *(v2f*)(zbase + ((size_t)i2 * FOUT + t * 16 + n) * 2) = p;
    }
}

// ---------------------------------------------------------------------------
// Block-CSR SpMM + bias (+ ReLU + LayerNorm), z in pair-interleaved layout:
//   out[r*16.., :] = LN(relu(sum_k bvals[r,k](16x16) @ z[bcols[r,k]*16.., :] + bias))
// One wave owns one block row (16 rows x F cols); LN row reduction is done
// with shfl_xor inside each 16-lane half (masks 1,2,4,8 keep bit 4 fixed).
// Output h / final logits are stored row-major (consumed as GEMM A / d_out).
// ---------------------------------------------------------------------------
template <int F, bool DO_LN>
__launch_bounds__(128) __global__
void spmm_kernel(const float* __restrict__ bvals, const int* __restrict__ bcols,
                 const float* __restrict__ zint, const float* __restrict__ bias,
                 const float* __restrict__ g, const float* __restrict__ beta,
                 float* __restrict__ out) {
  constexpr int NT = F / 16;
  const int lane = threadIdx.x & 31;
  const int wave = threadIdx.x >> 5;
  const int r  = blockIdx.x * (blockDim.x >> 5) + wave;  // block row
  const int n  = lane & 15;
  const int kk = lane >> 4;

  v8f acc[NT];
#pragma unroll
  for (int t = 0; t < NT; ++t)
#pragma unroll
    for (int i = 0; i < 8; ++i) acc[t][i] = 0.0f;

  const float* bv = bvals + (size_t)r * (GCN_K * GCN_T * GCN_T);
#pragma unroll
  for (int k = 0; k < GCN_K; ++k) {
    const int col = bcols[r * GCN_K + k];
    const float* av = bv + k * (GCN_T * GCN_T) + n * GCN_T; // A row m = lane%16
#pragma unroll
    for (int kb = 0; kb < GCN_T; kb += 4) {
      v2f a = *(const v2f*)(av + kb + 2 * kk);
      // z rows (col*16 + kb+2*kk, +1): one b64 from interleaved layout
      const float* z0 = zint + ((size_t)col * 8 + ((kb + 2 * kk) >> 1)) * F * 2;
#pragma unroll
      for (int t = 0; t < NT; ++t) {
        v2f b = *(const v2f*)(z0 + (t * 16 + n) * 2);
        acc[t] = wmma_f32(a, b, acc[t]);
      }
    }
  }

  float* orow = out + (size_t)(r * 16 + 8 * kk) * F;

  if (DO_LN) {
    // bias + relu in place, accumulate per-row sum / sumsq
    float s1[8], s2[8];
#pragma unroll
    for (int i = 0; i < 8; ++i) { s1[i] = 0.0f; s2[i] = 0.0f; }
#pragma unroll
    for (int t = 0; t < NT; ++t) {
      const float bc = bias[t * 16 + n];
#pragma unroll
      for (int i = 0; i < 8; ++i) {
        float v = acc[t][i] + bc;
        v = v > 0.0f ? v : 0.0f;
        acc[t][i] = v;
        s1[i] += v;
        s2[i] += v * v;
      }
    }
#pragma unroll
    for (int m = 1; m < 16; m <<= 1) {
#pragma unroll
      for (int i = 0; i < 8; ++i) {
        s1[i] += __shfl_xor(s1[i], m, 32);
        s2[i] += __shfl_xor(s2[i], m, 32);
      }
    }
    float mu[8], rs[8];
#pragma unroll
    for (int i = 0; i < 8; ++i) {
      mu[i] = s1[i] * (1.0f / F);
      float var = s2[i] * (1.0f / F) - mu[i] * mu[i];
      rs[i] = rsqrtf(var + LN_EPS);
    }
#pragma unroll
    for (int t = 0; t < NT; ++t) {
      const float gc = g[t * 16 + n];
      const float bt = beta[t * 16 + n];
#pragma unroll
      for (int i = 0; i < 8; ++i)
        orow[(size_t)i * F + t * 16 + n] = gc * (acc[t][i] - mu[i]) * rs[i] + bt;
    }
  } else {
    // final layer: bias only
#pragma unroll
    for (int t = 0; t < NT; ++t) {
      const float bc = bias[t * 16 + n];
#pragma unroll
      for (int i = 0; i < 8; ++i)
        orow[(size_t)i * F + t * 16 + n] = acc[t][i] + bc;
    }
  }
}

// ---------------------------------------------------------------------------
// Host side: 3 interleave pre-passes + 6 chained compute launches on `stream`.
// Input order (from setup_inputs):
//   0 features [N,128] f32      1 block_vals [BR,K,16,16] f32
//   2 W0 [128,256]  3 b0 [256]  4 W1 [256,256]  5 b1 [256]
//   6 W2 [256,64]   7 b2 [64]   8 g0 [256]  9 beta0 [256]
//  10 g1 [256]  11 beta1 [256]  12 block_cols [BR,K] i32
// ---------------------------------------------------------------------------
extern "C" void kernel_launch(void* const* d_in, const int* in_sizes, int n_in,
                              void* d_out, int out_size, void* d_ws, size_t ws_size,
                              hipStream_t stream) {
  (void)in_sizes; (void)n_in; (void)out_size; (void)ws_size;

  const float* features = (const float*)d_in[0];
  const float* bvals    = (const float*)d_in[1];
  const float* W0 = (const float*)d_in[2];
  const float* b0 = (const float*)d_in[3];
  const float* W1 = (const float*)d_in[4];
  const float* b1 = (const float*)d_in[5];
  const float* W2 = (const float*)d_in[6];
  const float* b2 = (const float*)d_in[7];
  const float* g0    = (const float*)d_in[8];
  const float* beta0 = (const float*)d_in[9];
  const float* g1    = (const float*)d_in[10];
  const float* beta1 = (const float*)d_in[11];
  const int*   bcols = (const int*)d_in[12];
  float* out = (float*)d_out;

  float* zbuf = (float*)d_ws;                        // [N,256] interleaved z
  float* hbuf = zbuf + (size_t)GCN_N * 256;          // [N,256] row-major h
  float* wp0  = hbuf + (size_t)GCN_N * 256;          // 128*256 interleaved W0
  float* wp1  = wp0 + 128 * 256;                     // 256*256 interleaved W1
  float* wp2  = wp1 + 256 * 256;                     // 256*64  interleaved W2

  // interleave weights (tiny; runs once per call, deterministic)
  interleave_rows<<<(128 * 256) / 256, 256, 0, stream>>>(W0, wp0, 128, 256);
  interleave_rows<<<(256 * 256) / 256, 256, 0, stream>>>(W1, wp1, 256, 256);
  interleave_rows<<<(256 * 64)  / 256, 256, 0, stream>>>(W2, wp2, 256, 64);

  const int WPB = 4;                 // waves per workgroup (128 threads)
  dim3 blk(32 * WPB);
  dim3 grd(GCN_BR / WPB);            // 8192 tiles / 4 waves = 2048 WGs

  // layer 0
  gemm_kernel<128, 256><<<grd, blk, 0, stream>>>(features, wp0, zbuf);
  spmm_kernel<256, true><<<grd, blk, 0, stream>>>(bvals, bcols, zbuf, b0, g0, beta0, hbuf);
  // layer 1
  gemm_kernel<256, 256><<<grd, blk, 0, stream>>>(hbuf, wp1, zbuf);
  spmm_kernel<256, true><<<grd, blk, 0, stream>>>(bvals, bcols, zbuf, b1, g1, beta1, hbuf);
  // output layer
  gemm_kernel<256, 64><<<grd, blk, 0, stream>>>(hbuf, wp2, zbuf);
  spmm_kernel<64, false><<<grd, blk, 0, stream>>>(bvals, bcols, zbuf, b2, nullptr, nullptr, out);
}